// GraphPool_53309134078317
// MI455X (gfx1250) — compile-verified
//
#include <hip/hip_runtime.h>
#include <math.h>

typedef float v2f __attribute__((ext_vector_type(2)));
typedef float v8f __attribute__((ext_vector_type(8)));

// ---------------------------------------------------------------------------
// Kernel 1: per-node scores y[b,n] = X[b,n,:] . (kernel / ||kernel||)
// One wave32 computes 16 nodes via V_WMMA_F32_16X16X4_F32, marching K by 4.
// A (16x4 f32): lanes 0-15 -> M=lane, VGPR0=K0,VGPR1=K1; lanes 16-31 -> K2,K3.
// B (4x16 f32): broadcast w into every column -> D is column-uniform, so
// lane 0 holds y[M=0..7] in acc[0..7], lane 16 holds y[M=8..15].
// ---------------------------------------------------------------------------
__global__ void score_kernel(const float* __restrict__ X,
                             const float* __restrict__ kern,
                             float* __restrict__ y,
                             int B, int N, int F, int tilesPerB) {
  const int lane  = threadIdx.x & 31;
  const int wave  = threadIdx.x >> 5;
  const int gwave = blockIdx.x * (blockDim.x >> 5) + wave;
  if (gwave >= B * tilesPerB) return;          // wave-uniform exit (EXEC stays all-ones for WMMA)
  const int b     = gwave / tilesPerB;
  const int t     = gwave % tilesPerB;
  const int node0 = t * 16;

  // ||kernel||^-1 : wave-cooperative sum of squares + wave32 xor reduction
  float s = 0.f;
  for (int i = lane; i < F; i += 32) { float v = kern[i]; s += v * v; }
  for (int off = 16; off > 0; off >>= 1) s += __shfl_xor(s, off, 32);
  const float inv = rsqrtf(s);

  const int m    = lane & 15;
  int row        = node0 + m;
  if (row >= N) row = N - 1;                   // clamp tail reads; writes guarded below
  const float* xrow = X + ((size_t)b * N + row) * F;
  const int koff    = (lane < 16) ? 0 : 2;

  v8f acc = {};
  for (int kk = 0; kk < F; kk += 4) {
    v2f a;  a.x  = xrow[kk + koff];           a.y  = xrow[kk + koff + 1];
    v2f bb; bb.x = kern[kk + koff] * inv;     bb.y = kern[kk + koff + 1] * inv;
    acc = __builtin_amdgcn_wmma_f32_16x16x4_f32(
        /*neg_a=*/false, a, /*neg_b=*/false, bb,
        /*c_mod=*/(short)0, acc, /*reuse_a=*/false, /*reuse_b=*/false);
  }

  if (lane == 0 || lane == 16) {
    const int mbase = node0 + ((lane == 0) ? 0 : 8);
#pragma unroll
    for (int r = 0; r < 8; ++r) {
      const int n = mbase + r;
      if (n < N) y[(size_t)b * N + n] = acc[r];
    }
  }
}

// ---------------------------------------------------------------------------
// Kernel 2: exact sorted top-k per batch. Bitonic sort of 4096 padded pairs
// in LDS (32 KB of the 320 KB WGP pool). Order: value desc, tie -> index asc
// (matches jax.lax.top_k). One 1024-thread block per batch.
// ---------------------------------------------------------------------------
__global__ void topk_kernel(const float* __restrict__ y,
                            int* __restrict__ wsidx,
                            int* __restrict__ outidx,
                            int N, int K) {
  constexpr int M = 4096;
  __shared__ float sv[M];
  __shared__ int   si[M];
  const int b = blockIdx.x;

  for (int i = threadIdx.x; i < M; i += blockDim.x) {
    sv[i] = (i < N) ? y[(size_t)b * N + i] : -__builtin_huge_valf();
    si[i] = i;
  }
  __syncthreads();

  for (int size = 2; size <= M; size <<= 1) {
    for (int stride = size >> 1; stride > 0; stride >>= 1) {
      for (int j = threadIdx.x; j < (M >> 1); j += blockDim.x) {
        const int pos     = 2 * j - (j & (stride - 1));
        const int partner = pos + stride;
        const bool up     = ((j & (size >> 1)) == 0);
        float va = sv[pos], vb = sv[partner];
        int   ia = si[pos], ib = si[partner];
        const bool aFirst = (va > vb) || (va == vb && ia < ib);
        if (up ? !aFirst : aFirst) {
          sv[pos] = vb; sv[partner] = va;
          si[pos] = ib; si[partner] = ia;
        }
      }
      __syncthreads();
    }
  }

  for (int i = threadIdx.x; i < K; i += blockDim.x) {
    const int v = si[i];
    wsidx[(size_t)b * K + i]  = v;
    outidx[(size_t)b * K + i] = v;
  }
}

// ---------------------------------------------------------------------------
// Kernel 3: new_X[b,i,:] = X[b, idx[b,i], :] * tanh(y[b, idx[b,i]])
// ---------------------------------------------------------------------------
__global__ void gatherX_kernel(const float* __restrict__ X,
                               const float* __restrict__ y,
                               const int* __restrict__ idx,
                               float* __restrict__ outX,
                               int N, int F, int K) {
  const int b = blockIdx.y, i = blockIdx.x;
  const int row = idx[(size_t)b * K + i];
  const float g = tanhf(y[(size_t)b * N + row]);
  const float* src = X + ((size_t)b * N + row) * F;
  float* dst = outX + ((size_t)b * K + i) * F;
  for (int f = threadIdx.x; f < F; f += blockDim.x) dst[f] = src[f] * g;
}

// ---------------------------------------------------------------------------
// Kernel 4 (bandwidth-dominant): new_A[b,i,j] = A[b, idx[i], idx[j]]
// idx row staged in LDS; 4 output rows per block; contiguous stores,
// scattered (L2-friendly, density-0.5) loads.
// ---------------------------------------------------------------------------
__global__ void gatherA_kernel(const float* __restrict__ A,
                               const int* __restrict__ idx,
                               float* __restrict__ outA,
                               int N, int K) {
  extern __shared__ int sidx[];                // K ints
  const int b  = blockIdx.y;
  const int i0 = blockIdx.x * 4;
  for (int j = threadIdx.x; j < K; j += blockDim.x)
    sidx[j] = idx[(size_t)b * K + j];
  __syncthreads();
#pragma unroll
  for (int ii = 0; ii < 4; ++ii) {
    const int i = i0 + ii;
    if (i >= K) break;
    const int row = sidx[i];
    const float* src = A + ((size_t)b * N + row) * N;
    float* dst = outA + ((size_t)b * K + i) * K;
    for (int j = threadIdx.x; j < K; j += blockDim.x) dst[j] = src[sidx[j]];
  }
}

// ---------------------------------------------------------------------------
extern "C" void kernel_launch(void* const* d_in, const int* in_sizes, int n_in,
                              void* d_out, int out_size, void* d_ws, size_t ws_size,
                              hipStream_t stream) {
  const float* X    = (const float*)d_in[0];
  const float* A    = (const float*)d_in[1];
  const float* kern = (const float*)d_in[2];

  // Derive shapes: in_sizes = {B*N*F, B*N*N, F, 1}; out_size = B*k*(F+k+1)
  const long long xN = (long long)in_sizes[0];
  const long long aN = (long long)in_sizes[1];
  const int F = in_sizes[2];
  const int N = (int)((aN / xN) * F);
  const int B = (int)(xN / ((long long)N * F));
  const long long per = (long long)out_size / B;        // k^2 + (F+1)k
  const double kd = (-(double)(F + 1) +
                     sqrt((double)(F + 1) * (F + 1) + 4.0 * (double)per)) * 0.5;
  const int K = (int)(kd + 0.5);

  float* ws_y  = (float*)d_ws;                           // [B*N]
  int*   ws_ix = (int*)(ws_y + (size_t)B * N);           // [B*K]

  float* out_newX = (float*)d_out;                       // [B,K,F]
  float* out_newA = out_newX + (size_t)B * K * F;        // [B,K,K]
  int*   out_idx  = (int*)(out_newA + (size_t)B * K * K);// [B,K,1] (int32 bits)

  // 1) scores via WMMA
  const int tilesPerB  = (N + 15) / 16;
  const int totalWaves = B * tilesPerB;
  const int wpb = 4;                                     // 128 threads = 4 waves
  const int blocks = (totalWaves + wpb - 1) / wpb;
  score_kernel<<<blocks, wpb * 32, 0, stream>>>(X, kern, ws_y, B, N, F, tilesPerB);

  // 2) sorted top-k per batch
  topk_kernel<<<B, 1024, 0, stream>>>(ws_y, ws_ix, out_idx, N, K);

  // 3) gated feature gather
  gatherX_kernel<<<dim3(K, B), 128, 0, stream>>>(X, ws_y, ws_ix, out_newX, N, F, K);

  // 4) adjacency row+col gather (dominant ~216 MB of HBM traffic)
  gatherA_kernel<<<dim3((K + 3) / 4, B), 256, (size_t)K * sizeof(int), stream>>>(
      A, ws_ix, out_newA, N, K);
}